// LieGen_33973191312037
// MI455X (gfx1250) — compile-verified
//
#include <hip/hip_runtime.h>

typedef __attribute__((ext_vector_type(2))) float v2f;
typedef __attribute__((ext_vector_type(8))) float v8f;

namespace {
constexpr int kB      = 65536;  // batches
constexpr int kD      = 32;     // matrix dim
constexpr int kK      = 4;      // generators
constexpr int kStride = 34;     // padded LDS row stride (floats): bank-spread, 8B-aligned pairs
constexpr int kWaves  = 4;      // waves (batches) per block
constexpr int kScaleS = 5;      // squarings  -> scale X = M / 32
constexpr int kTaylor = 8;      // Taylor/Horner degree
}

// 32x32x32 f32 matmul for one wave: C = scale * (A @ B) (+ I), operands in LDS.
// V_WMMA_F32_16X16X4_F32 tiles: 2x2 C-tiles, 8 K-chunks of 4 -> 32 WMMA ops.
// Epilogue is fused into the C-tile register->LDS store (no extra LDS round-trip).
__device__ __forceinline__ void mm32(const float* Am, const float* Bm, float* Cm,
                                     int lane, float scale, bool addIdent) {
  const int m = lane & 15;   // row (A) / col (B,C) within 16-tile
  const int g = lane >> 4;   // half-wave group selects K pair / C row half
  const int dv = m - 8 * g;  // diagonal hits when v == dv (within ti==tj tiles)
#pragma unroll
  for (int ti = 0; ti < 2; ++ti) {
#pragma unroll
    for (int tj = 0; tj < 2; ++tj) {
      v8f acc = {0.f, 0.f, 0.f, 0.f, 0.f, 0.f, 0.f, 0.f};
#pragma unroll
      for (int kk = 0; kk < 8; ++kk) {
        const int k0 = kk * 4 + 2 * g;                 // even -> 8B aligned pair
        // A fragment (16x4): lane holds (m, k0),(m, k0+1), contiguous in LDS
        v2f a = *(const v2f*)(Am + (ti * 16 + m) * kStride + k0);
        // B fragment (4x16): lane holds (k0, n),(k0+1, n)
        v2f b;
        b.x = Bm[(k0    ) * kStride + tj * 16 + m];
        b.y = Bm[(k0 + 1) * kStride + tj * 16 + m];
        acc = __builtin_amdgcn_wmma_f32_16x16x4_f32(
            /*neg_a=*/false, a, /*neg_b=*/false, b,
            /*c_mod=*/(short)0, acc, /*reuse_a=*/false, /*reuse_b=*/false);
      }
      // Fused epilogue + C tile store: vgpr v holds row (v + 8*g), col = m
#pragma unroll
      for (int v = 0; v < 8; ++v) {
        float r = acc[v] * scale;
        if (addIdent && (ti == tj)) r += (v == dv) ? 1.f : 0.f;
        Cm[(ti * 16 + v + 8 * g) * kStride + tj * 16 + m] = r;
      }
    }
  }
}

__global__ __launch_bounds__(kWaves * 32)
void lie_expm_apply(const float* __restrict__ z, const float* __restrict__ c,
                    const float* __restrict__ W, float* __restrict__ out) {
  __shared__ float Wl[kK * kD * kD];           // raw W, shared by all 4 waves
  __shared__ float Xs[kWaves][kD * kStride];   // X = M / 2^s
  __shared__ float Pa[kWaves][kD * kStride];   // ping
  __shared__ float Pb[kWaves][kD * kStride];   // pong

  const int tid  = threadIdx.x;
  const int wave = tid >> 5;
  const int lane = tid & 31;

  for (int i = tid; i < kK * kD * kD; i += kWaves * 32) Wl[i] = W[i];
  __syncthreads();

  const int batch = blockIdx.x * kWaves + wave;

  float cc[kK];
#pragma unroll
  for (int k = 0; k < kK; ++k) cc[k] = c[batch * kK + k];

  float* X = Xs[wave];
  float* P = Pa[wave];
  float* T = Pb[wave];

  // Build scaled generator sum: X = (1/2^s) * 0.5 * sum_k c_k (W_k - W_k^T);  P = I
  const float sc = 0.5f / (float)(1 << kScaleS);
  for (int j = 0; j < kD; ++j) {
    float m = 0.f;
#pragma unroll
    for (int k = 0; k < kK; ++k)
      m += cc[k] * (Wl[k * kD * kD + lane * kD + j] -
                    Wl[k * kD * kD + j * kD + lane]);
    X[lane * kStride + j] = m * sc;
    P[lane * kStride + j] = (lane == j) ? 1.f : 0.f;
  }
  asm volatile("s_wait_dscnt 0" ::: "memory");  // cross-lane LDS visibility in-wave

  // Horner (fully unrolled; 1/j folds to literal constants):
  //   P <- I + (X @ P) * (1/j),  j = d..1   => degree-8 Taylor of exp(X)
#pragma unroll
  for (int j = kTaylor; j >= 1; --j) {
    mm32(X, P, T, lane, 1.f / (float)j, /*addIdent=*/true);
    float* tmp = P; P = T; T = tmp;
  }

  // Squaring: P <- P @ P, s times  => exp(M)
  for (int i = 0; i < kScaleS; ++i) {
    mm32(P, P, T, lane, 1.f, /*addIdent=*/false);
    float* tmp = P; P = T; T = tmp;
  }

  // Apply: out_b = exp(M) @ z_b  (stage z into scratch row, per-lane dot)
  T[lane] = z[batch * kD + lane];
  asm volatile("s_wait_dscnt 0" ::: "memory");
  float acc = 0.f;
  for (int j = 0; j < kD; ++j) acc += P[lane * kStride + j] * T[j];
  out[batch * kD + lane] = acc;
}

extern "C" void kernel_launch(void* const* d_in, const int* in_sizes, int n_in,
                              void* d_out, int out_size, void* d_ws, size_t ws_size,
                              hipStream_t stream) {
  (void)in_sizes; (void)n_in; (void)out_size; (void)d_ws; (void)ws_size;
  const float* z = (const float*)d_in[0];
  const float* c = (const float*)d_in[1];
  const float* W = (const float*)d_in[2];
  float* out = (float*)d_out;
  dim3 grid(kB / kWaves);
  dim3 block(kWaves * 32);
  hipLaunchKernelGGL(lie_expm_apply, grid, block, 0, stream, z, c, W, out);
}